// LaceNat_59365037965618
// MI455X (gfx1250) — compile-verified
//
#include <hip/hip_runtime.h>
#include <hip/hip_bf16.h>
#include <stdint.h>

#define LOGZERO (-1e10f)

constexpr int TX    = 288;           // 9 x wave32; covers L=257 states
constexpr int LMAX  = 257;           // 2*128+1
constexpr int XMAXC = 1024;
constexpr int WORDS = XMAXC / 16;    // 64 packed 2-bit bp words per state
constexpr int PF    = 8;             // async gather pipeline depth (hides ~full HBM latency)

__global__ __launch_bounds__(TX, 1)
void lacenat_viterbi_kernel(const float* __restrict__ ctc,
                            const unsigned char* __restrict__ src_mask,
                            const int* __restrict__ src_size,
                            const int* __restrict__ ys,
                            const int* __restrict__ ylens,
                            const int* __restrict__ blank_p,
                            float* __restrict__ out,
                            int bs, int xmax, int vocab, int ymax)
{
    // ~88 KB static LDS (CDNA5: 320 KB per WGP)
    __shared__ unsigned      bpw[LMAX * WORDS];   // 2-bit packed backpointers
    __shared__ float         alpha[2][TX + 4];    // double-buffered DP row
    __shared__ float         lpbuf[PF][TX + 4];   // async gather ring
    __shared__ int           pathLds[LMAX + 3];
    __shared__ int           seqLds[XMAXC];
    __shared__ int           countsLds[XMAXC];
    __shared__ unsigned char maskLds[XMAXC];
    __shared__ float         endA[2];

    const int b      = blockIdx.x;
    const int tid    = threadIdx.x;
    const int blankv = blank_p[0];
    const int Lr     = 2 * ymax + 1;
    const int ssb    = src_size[b];
    const int pl     = 2 * ylens[b] + 1;
    const bool active = (tid < Lr);

    // ---- per-state constants ----
    int  path_l = blankv;
    bool block2 = true;                 // skip-edge (m2) blocked?
    if (active) {
        if (tid & 1) path_l = ys[(size_t)b * ymax + (tid >> 1)];
        if (tid >= 2) {
            int path_lm2 = (tid & 1) ? ys[(size_t)b * ymax + (tid >> 1) - 1] : blankv;
            block2 = (path_l == path_lm2);   // even states: blank==blank -> blocked
        }
    }
    const bool     outside = active && (tid >= pl);
    const unsigned voff    = active ? (unsigned)path_l * 4u : 0u;  // byte gather offset

    // ---- init shared ----
    if (active) {
        pathLds[tid]  = path_l;
        alpha[0][tid] = (tid == 0) ? 0.0f : LOGZERO;
    }
    for (int t = tid; t < xmax; t += TX) maskLds[t] = src_mask[(size_t)b * xmax + t];
    if (tid == 0) { endA[0] = LOGZERO; endA[1] = LOGZERO; }

    // ---- prime the async-to-LDS gather pipeline (ASYNCcnt-tracked, CDNA5) ----
    const size_t rowstride = (size_t)vocab;
    const float* base0     = ctc + (size_t)b * xmax * rowstride;
    for (int i = 0; i < PF; ++i) {
        unsigned lds_a = (unsigned)(size_t)&lpbuf[i][tid];         // low 32 bits == LDS offset
        unsigned long long gb = (unsigned long long)(base0 + (size_t)i * rowstride);
        asm volatile("global_load_async_to_lds_b32 %0, %1, %2"
                     :: "v"(lds_a), "v"(voff), "s"(gb) : "memory");
    }

    __syncthreads();

    // ---- DP over time: one barrier per step, gather latency hidden by ASYNCcnt ----
    unsigned bpword = 0;
    for (int t = 0; t < xmax; ++t) {
        if (t < xmax - (PF - 1)) {
            asm volatile("s_wait_asynccnt 7" ::: "memory");   // oldest slot landed
        } else {
            asm volatile("s_wait_asynccnt 0" ::: "memory");   // pipeline drain tail
        }

        const int cur  = t & 1;
        const int slot = t & (PF - 1);

        if (active) {
            float lp = maskLds[t] ? lpbuf[slot][tid] : LOGZERO;
            float c0 = alpha[cur][tid];
            float c1 = (tid >= 1) ? alpha[cur][tid - 1] : LOGZERO;
            float c2 = (tid >= 2 && !block2) ? alpha[cur][tid - 2] : LOGZERO;
            int bp; float mx;
            if (c0 >= c1 && c0 >= c2) { bp = 0; mx = c0; }   // jnp.argmax first-max order
            else if (c1 >= c2)        { bp = 1; mx = c1; }
            else                      { bp = 2; mx = c2; }
            if (outside) mx = LOGZERO;                       // mask AFTER argmax (bp kept)
            float an = mx + lp;
            alpha[cur ^ 1][tid] = an;
            bpword |= ((unsigned)bp) << (2 * (t & 15));
            if ((t & 15) == 15) { bpw[tid * WORDS + (t >> 4)] = bpword; bpword = 0; }
            if (t == ssb - 1) {                              // snapshot alpha[src_size]
                if (tid == pl - 1) endA[0] = an;
                if (tid == pl - 2) endA[1] = an;
            }
        }

        // refill the slot just consumed with the gather for step t+PF
        if (t + PF < xmax) {
            unsigned lds_a = (unsigned)(size_t)&lpbuf[slot][tid];
            unsigned long long gb = (unsigned long long)(base0 + (size_t)(t + PF) * rowstride);
            asm volatile("global_load_async_to_lds_b32 %0, %1, %2"
                         :: "v"(lds_a), "v"(voff), "s"(gb) : "memory");
        }
        __syncthreads();
    }

    // ---- sequential backtrace + dedup + cumsum, all from LDS (thread 0) ----
    if (tid == 0) {
        float s1 = endA[0], s2 = endA[1];
        int   start_state = (s1 > s2) ? (pl - 1) : (pl - 2);
        float score = fmaxf(s1, s2);
        int s = 0;
        for (int t = xmax - 1; t >= 0; --t) {
            if (t == ssb - 1) s = start_state;
            else if (t < ssb - 1) {
                int tt = t + 1;
                unsigned w = bpw[s * WORDS + (tt >> 4)];
                s -= (int)((w >> (2 * (tt & 15))) & 3u);
            } else s = 0;
            seqLds[t] = pathLds[s];
        }
        int c = 0, prev_orig = 0, prev_dd = 0;
        for (int t = 0; t < xmax; ++t) {
            int orig = seqLds[t];
            int dd   = (orig == prev_orig) ? 0 : orig;   // dedup vs ORIGINAL prev
            if (t > 0 && prev_dd != blankv) ++c;         // cumsum of shifted dedup != blank
            countsLds[t] = c;
            prev_orig = orig; prev_dd = dd;
        }
        size_t offY = (size_t)bs * (ymax + 2) * xmax;
        out[offY + b]      = (float)(ylens[b] + 1);      // ylen
        out[offY + bs + b] = score;                      // scores
    }
    __syncthreads();

    // ---- trigger_mask [b, ymax+2, xmax] as float 0/1, coalesced stores ----
    const int rows = ymax + 2;
    float* trig = out + (size_t)b * rows * xmax;
    for (int j = 0; j < rows; ++j) {
        const int  jj     = j - 1;
        const bool isSos  = (j == 0);
        const bool isLast = (jj == ymax);
        float* rowp = trig + (size_t)j * xmax;
        for (int t = tid; t < xmax; t += TX) {
            bool mv = maskLds[t] != 0;
            bool v;
            if (isSos)       v = (t == 0);
            else {
                v = (countsLds[t] == jj);
                if (isLast)  v = v || (t == ssb - 1);    // last-row override, then & mask
            }
            rowp[t] = (v && mv) ? 1.0f : 0.0f;
        }
    }
}

extern "C" void kernel_launch(void* const* d_in, const int* in_sizes, int n_in,
                              void* d_out, int out_size, void* d_ws, size_t ws_size,
                              hipStream_t stream) {
    const float*         ctc      = (const float*)d_in[0];
    const unsigned char* src_mask = (const unsigned char*)d_in[1];
    const int*           src_size = (const int*)d_in[2];
    const int*           ys       = (const int*)d_in[3];
    const int*           ylens    = (const int*)d_in[4];
    const int*           blank    = (const int*)d_in[5];

    int bs    = in_sizes[2];
    int xmax  = in_sizes[1] / bs;
    int ymax  = in_sizes[3] / bs;
    int vocab = (int)((long long)in_sizes[0] / ((long long)bs * (long long)xmax));

    lacenat_viterbi_kernel<<<bs, TX, 0, stream>>>(
        ctc, src_mask, src_size, ys, ylens, blank,
        (float*)d_out, bs, xmax, vocab, ymax);
}